// GSDRWLSNet_89721866813507
// MI455X (gfx1250) — compile-verified
//
#include <hip/hip_runtime.h>
#include <hip/hip_bf16.h>

typedef float v2f __attribute__((ext_vector_type(2)));
typedef float v8f __attribute__((ext_vector_type(8)));

#define BM 128
#define BN 64
#define BK 16
#define BKP (BK + 2)   // padded k-stride: gcd(18,64)=2 -> 16 distinct banks/half-wave

#define A_ELEMS ((BM * BK) / 256)   // 8 per thread
#define B_ELEMS ((BN * BK) / 256)   // 4 per thread

// D[grow*ldd + gcol*dcs] = scale*(A @ opB(B)) [+ cscale*Cin] [+ I] [+ *bias]
// A: row-major, lda, contiguous cols.
// B: if !TRANSB: logical B[k][n] = B[k*ldb + n*bcs]
//    if  TRANSB: logical B[k][n] = B[n*ldb + k*bcs]
template <bool TRANSB>
__global__ __launch_bounds__(256) void gemm_wmma_f32(
    const float* __restrict__ A, int lda,
    const float* __restrict__ B, int ldb, int bcs,
    float* __restrict__ D, long long ldd, int dcs,
    const float* __restrict__ Cin, int ldcin, float cscale,
    const float* __restrict__ bias,
    float scale, int addI, int M, int N, int K)
{
    // Both tiles keep K contiguous so every WMMA fragment is ONE aligned b64 LDS load.
    __shared__ float As[BM][BKP];
    __shared__ float Bs[BN][BKP];

    const int tid  = threadIdx.x;
    const int lane = tid & 31;
    const int wave = tid >> 5;
    const int wm   = wave >> 1;   // 0..3 -> M subtile
    const int wn   = wave & 1;    // 0..1 -> N subtile
    const int gm0  = blockIdx.y * BM;
    const int gn0  = blockIdx.x * BN;

    const v8f zero8 = {0.f, 0.f, 0.f, 0.f, 0.f, 0.f, 0.f, 0.f};
    v8f acc[2][2];
    acc[0][0] = zero8; acc[0][1] = zero8; acc[1][0] = zero8; acc[1][1] = zero8;

    const int rsel  = lane >> 4;          // 0: lanes 0-15, 1: lanes 16-31
    const int khalf = rsel * 2;           // K-pair selector per WMMA f32 layout
    const int arow0 = wm * 32 + (lane & 15);
    const int bcol0 = wn * 32 + (lane & 15);

    // ---- register-staged, software-pipelined global->LDS copies ----
    float areg[A_ELEMS], breg[B_ELEMS];

    auto loadTiles = [&](int k0) {
        #pragma unroll
        for (int i = 0; i < A_ELEMS; ++i) {
            const int idx = tid + i * 256;
            const int m   = idx >> 4;
            const int kk  = idx & 15;
            areg[i] = A[(long long)(gm0 + m) * lda + (k0 + kk)];
        }
        #pragma unroll
        for (int i = 0; i < B_ELEMS; ++i) {
            const int idx = tid + i * 256;
            if (!TRANSB) {
                const int kk = idx >> 6;   // coalesced over n in global
                const int n  = idx & 63;
                breg[i] = B[(long long)(k0 + kk) * ldb + (long long)(gn0 + n) * bcs];
            } else {
                const int n  = idx >> 4;   // coalesced over k in global
                const int kk = idx & 15;
                breg[i] = B[(long long)(gn0 + n) * ldb + (long long)(k0 + kk) * bcs];
            }
        }
    };

    auto storeTiles = [&]() {
        #pragma unroll
        for (int i = 0; i < A_ELEMS; ++i) {
            const int idx = tid + i * 256;
            As[idx >> 4][idx & 15] = areg[i];
        }
        #pragma unroll
        for (int i = 0; i < B_ELEMS; ++i) {
            const int idx = tid + i * 256;
            if (!TRANSB) Bs[idx & 63][idx >> 6] = breg[i];
            else         Bs[idx >> 4][idx & 15] = breg[i];
        }
    };

    loadTiles(0);

    for (int k0 = 0; k0 < K; k0 += BK) {
        storeTiles();
        __syncthreads();

        // Issue next tile's global loads now; they complete during the WMMAs below.
        if (k0 + BK < K) loadTiles(k0 + BK);

        #pragma unroll
        for (int kk = 0; kk < BK; kk += 4) {
            // A 16x4 frag: lane l<16 -> row l, {K=kk,kk+1}; lane l+16 -> {K=kk+2,kk+3}
            v2f a0 = *(const v2f*)&As[arow0][kk + khalf];
            v2f a1 = *(const v2f*)&As[arow0 + 16][kk + khalf];
            // B 4x16 frag: mirrored layout, N across lanes, K-pair contiguous in LDS
            v2f b0 = *(const v2f*)&Bs[bcol0][kk + khalf];
            v2f b1 = *(const v2f*)&Bs[bcol0 + 16][kk + khalf];

            acc[0][0] = __builtin_amdgcn_wmma_f32_16x16x4_f32(false, a0, false, b0, (short)0, acc[0][0], false, false);
            acc[0][1] = __builtin_amdgcn_wmma_f32_16x16x4_f32(false, a0, false, b1, (short)0, acc[0][1], false, false);
            acc[1][0] = __builtin_amdgcn_wmma_f32_16x16x4_f32(false, a1, false, b0, (short)0, acc[1][0], false, false);
            acc[1][1] = __builtin_amdgcn_wmma_f32_16x16x4_f32(false, a1, false, b1, (short)0, acc[1][1], false, false);
        }
        __syncthreads();
    }

    // ---- epilogue ----
    const int col0   = lane & 15;
    const int rowoff = rsel * 8;   // C/D layout: VGPR r -> row r (lanes 0-15) / r+8 (lanes 16-31)
    const float bval = bias ? *bias : 0.0f;

    #pragma unroll
    for (int ti = 0; ti < 2; ++ti) {
        #pragma unroll
        for (int tj = 0; tj < 2; ++tj) {
            const int gcol = gn0 + wn * 32 + tj * 16 + col0;
            #pragma unroll
            for (int r = 0; r < 8; ++r) {
                const int grow = gm0 + wm * 32 + ti * 16 + rowoff + r;
                float v = scale * acc[ti][tj][r] + bval;
                if (Cin)  v += cscale * Cin[(long long)grow * ldcin + gcol];
                if (addI && grow == gcol) v += 1.0f;
                D[(long long)grow * ldd + (long long)gcol * dcs] = v;
            }
        }
    }
}

// Row softmax with diagonal masked to -inf. One block per row; matrices are
// contiguous so blockIdx.x spans all (mat, row) pairs: base = blockIdx.x * n.
__global__ __launch_bounds__(256) void softmax_rows(
    const float* __restrict__ S, float* __restrict__ A, int n)
{
    const int row = blockIdx.x % n;
    const long long base = (long long)blockIdx.x * n;
    __shared__ float red[256];
    const int tid = threadIdx.x;

    float mx = -__builtin_inff();
    for (int b = tid; b < n; b += 256) {
        float v = (b == row) ? -__builtin_inff() : S[base + b];
        mx = fmaxf(mx, v);
    }
    red[tid] = mx; __syncthreads();
    for (int s = 128; s > 0; s >>= 1) {
        if (tid < s) red[tid] = fmaxf(red[tid], red[tid + s]);
        __syncthreads();
    }
    mx = red[0]; __syncthreads();

    float sum = 0.f;
    for (int b = tid; b < n; b += 256) {
        float v = (b == row) ? 0.0f : __expf(S[base + b] - mx);
        sum += v;
        A[base + b] = v;
    }
    red[tid] = sum; __syncthreads();
    for (int s = 128; s > 0; s >>= 1) {
        if (tid < s) red[tid] += red[tid + s];
        __syncthreads();
    }
    const float inv = 1.0f / red[0];
    for (int b = tid; b < n; b += 256) A[base + b] *= inv;
}

// X0 = I + alpha*A for all matrices (first Neumann/Horner term).
__global__ __launch_bounds__(256) void init_ppr(
    const float* __restrict__ Amats, float* __restrict__ X, int n, int nmats)
{
    const long long nn = (long long)n * n;
    const long long total = nmats * nn;
    for (long long i = (long long)blockIdx.x * 256 + threadIdx.x; i < total;
         i += (long long)gridDim.x * 256) {
        const long long w = i % nn;
        const int r = (int)(w / n), c = (int)(w % n);
        X[i] = 0.1f * Amats[i] + ((r == c) ? 1.0f : 0.0f);
    }
}

extern "C" void kernel_launch(void* const* d_in, const int* in_sizes, int n_in,
                              void* d_out, int out_size, void* d_ws, size_t ws_size,
                              hipStream_t stream) {
    (void)in_sizes; (void)n_in; (void)out_size; (void)ws_size;
    const float* x    = (const float*)d_in[0];   // [1024,1024]
    const float* Wb   = (const float*)d_in[1];   // [1024,2048]
    const float* Wemb = (const float*)d_in[2];   // [4,512,512,2]
    const float* bemb = (const float*)d_in[3];   // [4,2]
    float* out = (float*)d_out;
    float* ws  = (float*)d_ws;

    const int N = 1024, J = 4, BLK = 512;
    const long long NN = (long long)N * N;

    float* feats = ws;                               // 1024*2048
    float* Tbuf  = feats + (size_t)N * 2048;         // 1024*512 (reused per (j,k))
    float* Sbuf  = Tbuf + (size_t)N * BLK;           // 8 * NN
    float* Abuf  = Sbuf + 8 * NN;                    // 8 * NN
    float* Xa    = Abuf + 8 * NN;                    // 8 * NN
    float* Xb    = Xa   + 8 * NN;                    // 8 * NN

    const dim3 blk(256);

    // feats = x @ Wb   [1024 x 2048, K=1024]
    gemm_wmma_f32<false><<<dim3(2048 / BN, N / BM), blk, 0, stream>>>(
        x, 1024, Wb, 2048, 1, feats, 2048, 1, nullptr, 0, 0.f, nullptr,
        1.0f, 0, N, 2048, 1024);

    // S_jk = (xb_j @ W_jk) @ xb_j^T + b_jk
    for (int j = 0; j < J; ++j) {
        const float* xbj = feats + (size_t)j * BLK;          // lda = 2048
        for (int k = 0; k < 2; ++k) {
            const int m = j * 2 + k;
            const float* Wjk = Wemb + (size_t)j * BLK * BLK * 2 + k; // row stride 1024, col stride 2
            gemm_wmma_f32<false><<<dim3(BLK / BN, N / BM), blk, 0, stream>>>(
                xbj, 2048, Wjk, 1024, 2, Tbuf, BLK, 1, nullptr, 0, 0.f, nullptr,
                1.0f, 0, N, BLK, BLK);
            gemm_wmma_f32<true><<<dim3(N / BN, N / BM), blk, 0, stream>>>(
                Tbuf, BLK, xbj, 2048, 1, Sbuf + m * NN, N, 1, nullptr, 0, 0.f,
                bemb + j * 2 + k, 1.0f, 0, N, N, BLK);
        }
    }

    // A_jk = softmax_rows(S_jk with diag = -inf)   (bias cancels; harmless)
    softmax_rows<<<8 * N, blk, 0, stream>>>(Sbuf, Abuf, N);

    // Neumann/Horner for (I - alpha*A)^{-1}: X0 = I + aA; X <- I + aA*X (x9)
    // ||aA||_inf = 0.1 (rows stochastic) -> truncation ~1e-11, below fp32 eps.
    init_ppr<<<2048, blk, 0, stream>>>(Abuf, Xa, N, 8);
    float* Xsrc = Xa;
    float* Xdst = Xb;
    for (int it = 0; it < 9; ++it) {
        for (int m = 0; m < 8; ++m) {
            gemm_wmma_f32<false><<<dim3(N / BN, N / BM), blk, 0, stream>>>(
                Abuf + m * NN, N, Xsrc + m * NN, N, 1, Xdst + m * NN, N, 1,
                nullptr, 0, 0.f, nullptr, 0.1f, 1, N, N, N);
        }
        float* t = Xsrc; Xsrc = Xdst; Xdst = t;
    }
    // Xsrc now holds (I - alpha*A)^{-1}; fold (1-alpha) into the ana GEMM scale.

    // one_jik = 0.1 * S_jk @ A_ik^T + 0.9 * S_jk
    // ana_jik = 0.9 * S_jk @ X_ik^T
    // out flat: ((j*4+i)*(2*NN) + p)*2 + k ; p = a*N + c (one), NN + a*N + c (ana)
    for (int j = 0; j < J; ++j) {
        for (int i = 0; i < J; ++i) {
            for (int k = 0; k < 2; ++k) {
                const int mjk = j * 2 + k;
                const int mik = i * 2 + k;
                float* base = out + ((size_t)(j * 4 + i) * 2 * NN) * 2 + k;
                gemm_wmma_f32<true><<<dim3(N / BN, N / BM), blk, 0, stream>>>(
                    Sbuf + mjk * NN, N, Abuf + mik * NN, N, 1,
                    base, 2048, 2, Sbuf + mjk * NN, N, 0.9f, nullptr,
                    0.1f, 0, N, N, N);
                gemm_wmma_f32<true><<<dim3(N / BN, N / BM), blk, 0, stream>>>(
                    Sbuf + mjk * NN, N, Xsrc + mik * NN, N, 1,
                    base + NN * 2, 2048, 2, nullptr, 0, 0.f, nullptr,
                    0.9f, 0, N, N, N);
            }
        }
    }
}